// TemporalPropagator_6932077216157
// MI455X (gfx1250) — compile-verified
//
#include <hip/hip_runtime.h>
#include <math.h>

#define DD  64
#define TT  16
#define HWN 4096
#define BBN 2

typedef float v2f __attribute__((ext_vector_type(2)));
typedef float v8f __attribute__((ext_vector_type(8)));

// ---------------- workspace layout (float offsets) ----------------
enum : int {
  OFF_WRE   = 0,      // w plane re (64x64, row = input dim)
  OFF_WIM   = 4096,
  OFF_VRE   = 8192,   // w_inv plane re
  OFF_VIM   = 12288,
  OFF_RAWRE = 16384,
  OFF_RAWIM = 20480,
  OFF_PRE   = 24576,  // perturb
  OFF_PIM   = 28672,
  OFF_TARE  = 32768,  // Neumann ping
  OFF_TAIM  = 36864,
  OFF_TBRE  = 40960,  // Neumann pong
  OFF_TBIM  = 45056,
  OFF_INVRE = 49152,
  OFF_INVIM = 53248,
  OFF_FWRE  = 57344,  // fragment-packed w_re   (64 frags * 64 floats)
  OFF_FWIM  = 61440,
  OFF_FVRE  = 65536,
  OFF_FVIM  = 69632,
  OFF_FU1   = 73728,  // unc_w1 frags (16 frags * 64)
  OFF_FU2   = 74752,  // unc_w2 frags (16 frags * 64)
  OFF_XM    = 75776,  // x spatial mean  (B*T*64)
  OFF_DRE   = 77824,  // decay re       (B*T*64)
  OFF_DIMG  = 79872,
  OFF_FRE   = 81920,  // forcing re
  OFF_FIMG  = 83968,
  OFF_DFRE  = 86016,  // flux decay
  OFF_DFIM  = 88064,
  OFF_FFRE  = 90112,  // flux forcing
  OFF_FFIM  = 92160,
  OFF_GSRE  = 94208,  // gate*source
  OFF_GSIM  = 96256,
  OFF_DTR   = 98304,
  WS_FLOATS = 98368
};

static __device__ __forceinline__ float softplusf(float x) {
  return (x > 20.f) ? x : log1pf(expf(x));
}
static __device__ __forceinline__ float sigmoidf(float x) {
  return 1.f / (1.f + expf(-x));
}
static __device__ __forceinline__ v8f v8zero() {
  v8f z; for (int i = 0; i < 8; ++i) z[i] = 0.f; return z;
}
// V_WMMA_F32_16X16X4_F32 : D(16x16 f32) = A(16x4 f32) * B(4x16 f32) + C
static __device__ __forceinline__ v8f wmma4(v2f a, v2f b, v8f c) {
  return __builtin_amdgcn_wmma_f32_16x16x4_f32(false, a, false, b, (short)0, c,
                                               false, false);
}
static __device__ __forceinline__ void gsync() {
  __threadfence();
  __syncthreads();
}
// CDNA5 async global->LDS copy (ASYNCcnt-tracked), 16B per lane, coalesced.
static __device__ __forceinline__ void async_load_b128(unsigned lds_off,
                                                       const void* gaddr) {
  asm volatile("global_load_async_to_lds_b128 %0, %1, off"
               :: "v"(lds_off), "v"(gaddr)
               : "memory");
}

// Pack a KxN row-major plane into WMMA B-fragment order:
// frag f = (k0/4)*NT + nt, lane l, comps {0,1};  element = plane[(k0+2*(l>>4)+c)*Nfull + nt*16+(l&15)]
static __device__ void write_frags(const float* __restrict__ plane,
                                   float* __restrict__ frag,
                                   int K, int Nfull, int NT, int tid) {
  const int total = (K / 4) * NT * 64;
  for (int i = tid; i < total; i += 256) {
    int f = i >> 6;
    int r = i & 63;
    int l = r >> 1;
    int comp = r & 1;
    int k0 = (f / NT) * 4;
    int nt = f - (f / NT) * NT;
    int k = k0 + 2 * (l >> 4) + comp;
    int n = nt * 16 + (l & 15);
    frag[i] = plane[k * Nfull + n];
  }
}

// ================= K1: spatial mean of x over (H,W) =================
__global__ __launch_bounds__(256)
void k1_xmean(const float* __restrict__ x, float* __restrict__ ws) {
  __shared__ float red[256];
  const int bt = blockIdx.x;
  const int tid = threadIdx.x;
  const int ch = tid & 63;
  const int grp = tid >> 6;
  const float* base = x + (size_t)bt * HWN * DD;
  float acc = 0.f;
  for (int p = grp * 1024; p < grp * 1024 + 1024; ++p)
    acc += base[(size_t)p * DD + ch];
  red[tid] = acc;
  __syncthreads();
  if (tid < 64) {
    float s = red[tid] + red[tid + 64] + red[tid + 128] + red[tid + 192];
    ws[OFF_XM + bt * 64 + tid] = s * (1.f / 4096.f);
  }
}

// ================= K0: matrices + per-(b,t) decay/forcing =================
__global__ __launch_bounds__(256)
void k0_prepare(const float* __restrict__ w_re_in, const float* __restrict__ w_im_in,
                const float* __restrict__ alpha_logit,
                const float* __restrict__ lam_re, const float* __restrict__ lam_im,
                const float* __restrict__ ftd_re, const float* __restrict__ ftd_im,
                const float* __restrict__ dt_seq,
                const float* __restrict__ unc_w1, const float* __restrict__ unc_w2,
                float* __restrict__ ws) {
  const int tid = threadIdx.x;
  float* rawR = ws + OFF_RAWRE; float* rawI = ws + OFF_RAWIM;
  float* pR = ws + OFF_PRE;     float* pI = ws + OFF_PIM;
  float* invR = ws + OFF_INVRE; float* invI = ws + OFF_INVIM;
  __shared__ float vre[64], vim[64], ure[64], uim[64], red[64];
  __shared__ float sh_sigma, sh_rn, sh_bscale, sh_ascale;

  // raw = (tanh(w_re) + i tanh(w_im)) / sqrt(64)
  for (int i = tid; i < 4096; i += 256) {
    rawR[i] = tanhf(w_re_in[i]) * 0.125f;
    rawI[i] = tanhf(w_im_in[i]) * 0.125f;
  }
  gsync();

  // power iteration: sigma_max(raw)
  if (tid < 64) { vre[tid] = 0.125f; vim[tid] = 0.f; }
  __syncthreads();
  for (int it = 0; it < 32; ++it) {
    if (tid < 64) {
      float sr = 0.f, si = 0.f;
      for (int k = 0; k < 64; ++k) {
        float ar = rawR[tid * 64 + k], ai = rawI[tid * 64 + k];
        float cr = vre[k], ci = vim[k];
        sr += ar * cr - ai * ci;
        si += ar * ci + ai * cr;
      }
      ure[tid] = sr; uim[tid] = si; red[tid] = sr * sr + si * si;
    }
    __syncthreads();
    if (tid == 0) {
      float s = 0.f; for (int k = 0; k < 64; ++k) s += red[k];
      sh_sigma = sqrtf(s);
    }
    __syncthreads();
    if (tid < 64) {
      float sr = 0.f, si = 0.f;
      for (int r = 0; r < 64; ++r) {
        float ar = rawR[r * 64 + tid], ai = rawI[r * 64 + tid]; // conj(raw)
        float cr = ure[r], ci = uim[r];
        sr += ar * cr + ai * ci;
        si += ar * ci - ai * cr;
      }
      vre[tid] = sr; vim[tid] = si; red[tid] = sr * sr + si * si;
    }
    __syncthreads();
    if (tid == 0) {
      float s = 0.f; for (int k = 0; k < 64; ++k) s += red[k];
      sh_rn = rsqrtf(fmaxf(s, 1e-30f));
    }
    __syncthreads();
    if (tid < 64) { vre[tid] *= sh_rn; vim[tid] *= sh_rn; }
    __syncthreads();
  }
  if (tid == 0) {
    float alpha = sigmoidf(alpha_logit[0]);
    float sn = fmaxf(sh_sigma, 1.1920929e-7f);
    float scale = fminf(0.5f / sn, 1.f);
    sh_bscale = (1.f - alpha) * scale;
    sh_ascale = 1.f + alpha * 0.001f;
  }
  __syncthreads();
  const float bscale = sh_bscale;
  const float ascale = sh_ascale;

  // perturb; Neumann init
  float* taR = ws + OFF_TARE; float* taI = ws + OFF_TAIM;
  float* tbR = ws + OFF_TBRE; float* tbI = ws + OFF_TBIM;
  for (int i = tid; i < 4096; i += 256) {
    pR[i] = bscale * rawR[i];
    pI[i] = bscale * rawI[i];
    float e = ((i >> 6) == (i & 63)) ? 1.f : 0.f;
    invR[i] = e; invI[i] = 0.f;
    taR[i] = e; taI[i] = 0.f;
  }
  gsync();
  for (int it = 0; it < 8; ++it) {
    const float* sR = (it & 1) ? tbR : taR; const float* sI = (it & 1) ? tbI : taI;
    float* dR = (it & 1) ? taR : tbR;       float* dI = (it & 1) ? taI : tbI;
    for (int i = tid; i < 4096; i += 256) {
      int r = i >> 6, c = i & 63;
      float ar = 0.f, ai = 0.f;
      for (int k = 0; k < 64; ++k) {
        float tr = sR[r * 64 + k], ti = sI[r * 64 + k];
        float qr = pR[k * 64 + c], qi = pI[k * 64 + c];
        ar += tr * qr - ti * qi;
        ai += tr * qi + ti * qr;
      }
      ar = -ar; ai = -ai;
      dR[i] = ar; dI[i] = ai;
      invR[i] += ar; invI[i] += ai;
    }
    gsync();
  }

  // w = ascale * (dft @ (I + perturb)) ; w_inv = inv @ dft^H / ascale
  const float step = 6.283185307179586f / 64.f;
  for (int i = tid; i < 4096; i += 256) {
    int n = i >> 6, c = i & 63;
    float wr = 0.f, wi = 0.f;
    for (int k = 0; k < 64; ++k) {
      int nk = (n * k) & 63;
      float th = step * (float)nk;
      float cc = cosf(th), ss = sinf(th);            // dft = (cc - i ss)/8
      float lr = ((k == c) ? 1.f : 0.f) + pR[k * 64 + c];
      float li = pI[k * 64 + c];
      wr += cc * lr + ss * li;
      wi += cc * li - ss * lr;
    }
    ws[OFF_WRE + i] = wr * 0.125f * ascale;
    ws[OFF_WIM + i] = wi * 0.125f * ascale;
    float vr = 0.f, vi = 0.f;
    for (int k = 0; k < 64; ++k) {
      int ck = (c * k) & 63;
      float th = step * (float)ck;
      float cc = cosf(th), ss = sinf(th);            // conj(dft[c][k]) = (cc + i ss)/8
      float ir = invR[n * 64 + k], ii = invI[n * 64 + k];
      vr += ir * cc - ii * ss;
      vi += ir * ss + ii * cc;
    }
    ws[OFF_VRE + i] = vr * 0.125f / ascale;
    ws[OFF_VIM + i] = vi * 0.125f / ascale;
  }
  gsync();

  write_frags(ws + OFF_WRE, ws + OFF_FWRE, 64, 64, 4, tid);
  write_frags(ws + OFF_WIM, ws + OFF_FWIM, 64, 64, 4, tid);
  write_frags(ws + OFF_VRE, ws + OFF_FVRE, 64, 64, 4, tid);
  write_frags(ws + OFF_VIM, ws + OFF_FVIM, 64, 64, 4, tid);
  write_frags(unc_w1, ws + OFF_FU1, 64, 16, 1, tid);
  write_frags(unc_w2, ws + OFF_FU2, 16, 64, 4, tid);

  if (tid < 32) ws[OFF_DTR + tid] = dt_seq[tid];  // DT_REF = 1.0

  // decay / forcing for both lam sets
  for (int i = tid; i < 2048; i += 256) {
    int bt = i >> 6, c = i & 63;
    float dtr = dt_seq[bt];
    for (int which = 0; which < 2; ++which) {
      float lr = -softplusf(which ? ftd_re[c] : lam_re[c]);
      float li = which ? ftd_im[c] : lam_im[c];
      float zr = lr * dtr, zi = li * dtr;
      float er = expf(zr);
      float dr = er * cosf(zi), di = er * sinf(zi);
      float m2 = zr * zr + zi * zi;
      float fr, fi;
      if (m2 > 1e-14f) {                     // |z| > 1e-7  -> phi1 = (e^z - 1)/z
        float nr = dr - 1.f, ni = di;
        fr = (nr * zr + ni * zi) / m2;
        fi = (ni * zr - nr * zi) / m2;
      } else {                               // taylor 1 + z/2 + z^2/6
        float z2r = zr * zr - zi * zi, z2i = 2.f * zr * zi;
        fr = 1.f + 0.5f * zr + z2r * (1.f / 6.f);
        fi = 0.5f * zi + z2i * (1.f / 6.f);
      }
      if (which == 0) {
        ws[OFF_DRE + i] = dr;  ws[OFF_DIMG + i] = di;
        ws[OFF_FRE + i] = dtr * fr;  ws[OFF_FIMG + i] = dtr * fi;
      } else {
        ws[OFF_DFRE + i] = dr; ws[OFF_DFIM + i] = di;
        ws[OFF_FFRE + i] = dtr * fr; ws[OFF_FFIM + i] = dtr * fi;
      }
    }
  }
}

// ================= K2: flux transformer (tiny, sequential over t) =================
__global__ __launch_bounds__(256)
void k2_flux(float* __restrict__ ws,
             const float* __restrict__ ft_in_w, const float* __restrict__ ft_in_b,
             const float* __restrict__ ft_out_w, const float* __restrict__ ft_out_b,
             const float* __restrict__ ft_gc_w, const float* __restrict__ ft_gc_b,
             const float* __restrict__ ft_gate_w, const float* __restrict__ ft_gate_b,
             const float* __restrict__ ft_h0_re, const float* __restrict__ ft_h0_im) {
  __shared__ float xc[2][128], xi[2][128], st[2][128], so[2][128], gc[2][128];
  const int tid = threadIdx.x;
  const int bbv = tid >> 7;   // batch handled by this half-block
  const int j = tid & 127;
  const float* wre = ws + OFF_WRE;
  const float* wim = ws + OFF_WIM;
  const float* xm = ws + OFF_XM;
  float fr = 0.f, fi = 0.f;
  if (j < 64) { fr = ft_h0_re[j]; fi = ft_h0_im[j]; }

  for (int t = 0; t < TT; ++t) {
    const int bt = bbv * TT + t;
    if (j < 64) {  // x_mean_c = mean(x) @ w  (linearity of mean & matmul)
      float sr = 0.f, si = 0.f;
      for (int k = 0; k < 64; ++k) {
        float xv = xm[bt * 64 + k];
        sr += xv * wre[k * 64 + j];
        si += xv * wim[k * 64 + j];
      }
      xc[bbv][j] = sr; xc[bbv][64 + j] = si;
    }
    __syncthreads();
    {
      float acc = ft_in_b[j];
      for (int k = 0; k < 128; ++k) acc += xc[bbv][k] * ft_in_w[k * 128 + j];
      xi[bbv][j] = acc;
    }
    __syncthreads();
    if (j < 64) {  // scan step
      float xr = xi[bbv][j], xiv = xi[bbv][64 + j];
      float ffr = ws[OFF_FFRE + bt * 64 + j], ffi = ws[OFF_FFIM + bt * 64 + j];
      float dfr = ws[OFF_DFRE + bt * 64 + j], dfi = ws[OFF_DFIM + bt * 64 + j];
      float ur = xr * ffr - xiv * ffi, ui = xr * ffi + xiv * ffr;
      float nr = fr * dfr - fi * dfi + ur;
      float ni = fr * dfi + fi * dfr + ui;
      fr = nr; fi = ni;
      st[bbv][j] = fr; st[bbv][64 + j] = fi;
    }
    __syncthreads();
    {
      float acc = ft_out_b[j];
      for (int k = 0; k < 128; ++k) acc += st[bbv][k] * ft_out_w[k * 128 + j];
      so[bbv][j] = acc;
    }
    __syncthreads();
    {
      float acc = ft_gc_b[j];
      for (int k = 0; k < 128; ++k) acc += st[bbv][k] * ft_gc_w[k * 128 + j];
      for (int k = 0; k < 128; ++k) acc += so[bbv][k] * ft_gc_w[(128 + k) * 128 + j];
      gc[bbv][j] = acc;
    }
    __syncthreads();
    if (j < 64) {
      float acc = ft_gate_b[j];
      for (int k = 0; k < 128; ++k) acc += gc[bbv][k] * ft_gate_w[k * 64 + j];
      float g = sigmoidf(acc) * 0.98f + 0.01f;
      ws[OFF_GSRE + bt * 64 + j] = g * so[bbv][j];
      ws[OFF_GSIM + bt * 64 + j] = g * so[bbv][64 + j];
    }
    __syncthreads();
  }
}

// ================= K3: fused encode -> scan -> decode -> uncertainty =================
#define STG 68            // padded LDS row stride (bank-conflict-free A-frag reads, 16B rows)
#define STG2 18           // padded stride for 16x16 hdn staging
#define XBUF (16 * STG)   // one 16x64 tile plane
#define WAVE_XLDS (2 * XBUF)   // double-buffered x tile
#define WAVE_HLDS (2 * XBUF)   // h re + im planes

__global__ __launch_bounds__(64)
void k3_main(const float* __restrict__ x, const float* __restrict__ ws,
             const float* __restrict__ h0_re, const float* __restrict__ h0_im,
             const float* __restrict__ unc_b1, const float* __restrict__ unc_b2,
             float* __restrict__ out) {
  __shared__ __align__(16) float xs[2 * WAVE_XLDS];  // 2 waves
  __shared__ __align__(16) float hs[2 * WAVE_HLDS];  // 2 waves
  const int tid = threadIdx.x;
  const int wv = tid >> 5;
  const int lane = tid & 31;
  const int gw = blockIdx.x * 2 + wv;
  const int b = gw >> 8;           // 256 strips per batch
  const int strip = gw & 255;
  const int row0 = strip * 16;
  const int lrow = lane & 15;      // A: row m / D: col n
  const int lhalf = lane >> 4;
  const int kb = 2 * lhalf;        // A/B fragment K sub-offset

  const float* fwre = ws + OFF_FWRE;
  const float* fwim = ws + OFF_FWIM;
  const float* fvre = ws + OFF_FVRE;
  const float* fvim = ws + OFF_FVIM;
  const float* fu1 = ws + OFF_FU1;
  const float* fu2 = ws + OFF_FU2;
  float* xw = xs + wv * WAVE_XLDS;
  float* st = hs + wv * WAVE_HLDS;

  // issue one 16x64 f32 tile (4KB) as 8 coalesced async b128 copies: 2 rows/op,
  // lane -> row 2i+lhalf, 16B segment lrow
  auto issue_tile = [&](int tload) {
    const float* gb = x + (((size_t)(b * TT + tload) * HWN) + row0) * DD;
    float* db = xw + (tload & 1) * XBUF;
#pragma unroll
    for (int i = 0; i < 8; ++i) {
      const float* g = gb + (2 * i + lhalf) * DD + (lrow << 2);
      unsigned loff =
          (unsigned)(uintptr_t)(db + (2 * i + lhalf) * STG + (lrow << 2));
      async_load_b128(loff, g);
    }
  };

  // h state in D-tile layout, kept in registers across the whole time scan
  v8f hre[4], him[4];
  for (int nt = 0; nt < 4; ++nt) {
    float hr0 = h0_re[nt * 16 + lrow];
    float hi0 = h0_im[nt * 16 + lrow];
    for (int r = 0; r < 8; ++r) { hre[nt][r] = hr0; him[nt][r] = hi0; }
  }

  issue_tile(0);

  for (int t = 0; t < TT; ++t) {
    // prefetch next tile while this one is consumed; then wait for current tile
    if (t + 1 < TT) {
      issue_tile(t + 1);
      asm volatile("s_wait_asynccnt 0x8" ::: "memory");
    } else {
      asm volatile("s_wait_asynccnt 0x0" ::: "memory");
    }

    // ---- read x tile from LDS as 16x4 f32 A-fragments
    const float* xcur = xw + (t & 1) * XBUF;
    v2f ax[16];
    for (int k0 = 0; k0 < 16; ++k0)
      ax[k0] = *(const v2f*)(xcur + lrow * STG + 4 * k0 + kb);

    // ---- encode: h_enc = x @ w (complex; x real -> two real WMMA chains)
    v8f er[4], ei[4];
    for (int nt = 0; nt < 4; ++nt) { er[nt] = v8zero(); ei[nt] = v8zero(); }
    for (int k0 = 0; k0 < 16; ++k0) {
      v2f a = ax[k0];
      for (int nt = 0; nt < 4; ++nt) {
        v2f br = *(const v2f*)(fwre + ((k0 * 4 + nt) << 6) + (lane << 1));
        v2f bi = *(const v2f*)(fwim + ((k0 * 4 + nt) << 6) + (lane << 1));
        er[nt] = wmma4(a, br, er[nt]);
        ei[nt] = wmma4(a, bi, ei[nt]);
      }
    }

    // ---- time-scan update: h = h*decay + (h_enc + gate*source)*forcing
    const int pb = (b * TT + t) * DD;
    for (int nt = 0; nt < 4; ++nt) {
      const int ch = nt * 16 + lrow;
      float dre = ws[OFF_DRE + pb + ch], dimv = ws[OFF_DIMG + pb + ch];
      float fre = ws[OFF_FRE + pb + ch], fimv = ws[OFF_FIMG + pb + ch];
      float gr = ws[OFF_GSRE + pb + ch], gi = ws[OFF_GSIM + pb + ch];
      v8f nr, ni;
      for (int r = 0; r < 8; ++r) {
        float ur = er[nt][r] + gr;
        float ui = ei[nt][r] + gi;
        nr[r] = hre[nt][r] * dre - him[nt][r] * dimv + ur * fre - ui * fimv;
        ni[r] = hre[nt][r] * dimv + him[nt][r] * dre + ur * fimv + ui * fre;
      }
      hre[nt] = nr; him[nt] = ni;
    }

    // ---- stage h (D layout) to LDS so it can be re-read as A fragments
    for (int nt = 0; nt < 4; ++nt) {
      const int ch = nt * 16 + lrow;
      for (int r = 0; r < 8; ++r) {
        int m = r + 8 * lhalf;
        st[m * STG + ch] = hre[nt][r];
        st[16 * STG + m * STG + ch] = him[nt][r];
      }
    }
    __syncthreads();

    v2f ar[16], ai2[16], aa[16];
    for (int k0 = 0; k0 < 16; ++k0) {
      ar[k0] = *(const v2f*)(st + lrow * STG + 4 * k0 + kb);
      ai2[k0] = *(const v2f*)(st + 16 * STG + lrow * STG + 4 * k0 + kb);
      v2f m;
      m.x = sqrtf(ar[k0].x * ar[k0].x + ai2[k0].x * ai2[k0].x);
      m.y = sqrtf(ar[k0].y * ar[k0].y + ai2[k0].y * ai2[k0].y);
      aa[k0] = m;
    }

    // ---- decode y = Re(h @ w_inv) and hdn = amp @ unc_w1
    v8f y[4];
    for (int nt = 0; nt < 4; ++nt) y[nt] = v8zero();
    v8f hd = v8zero();
    for (int k0 = 0; k0 < 16; ++k0) {
      v2f nim; nim.x = -ai2[k0].x; nim.y = -ai2[k0].y;
      for (int nt = 0; nt < 4; ++nt) {
        v2f br = *(const v2f*)(fvre + ((k0 * 4 + nt) << 6) + (lane << 1));
        v2f bi = *(const v2f*)(fvim + ((k0 * 4 + nt) << 6) + (lane << 1));
        y[nt] = wmma4(ar[k0], br, y[nt]);
        y[nt] = wmma4(nim, bi, y[nt]);   // - him @ winv_im
      }
      v2f b1 = *(const v2f*)(fu1 + (k0 << 6) + (lane << 1));
      hd = wmma4(aa[k0], b1, hd);
    }

    // silu(hdn + b1)
    {
      float b1v = unc_b1[lrow];
      for (int r = 0; r < 8; ++r) {
        float z = hd[r] + b1v;
        hd[r] = z * sigmoidf(z);
      }
    }
    __syncthreads();
    for (int r = 0; r < 8; ++r) {        // stage 16x16 hdn, D layout -> LDS
      int m = r + 8 * lhalf;
      st[m * STG2 + lrow] = hd[r];
    }
    __syncthreads();
    v2f ah[4];
    for (int k0 = 0; k0 < 4; ++k0)
      ah[k0] = *(const v2f*)(st + lrow * STG2 + 4 * k0 + kb);

    v8f sg[4];
    for (int nt = 0; nt < 4; ++nt) sg[nt] = v8zero();
    for (int k0 = 0; k0 < 4; ++k0)
      for (int nt = 0; nt < 4; ++nt) {
        v2f b2f = *(const v2f*)(fu2 + ((k0 * 4 + nt) << 6) + (lane << 1));
        sg[nt] = wmma4(ah[k0], b2f, sg[nt]);
      }

    // ---- sigma & output
    for (int nt = 0; nt < 4; ++nt) {
      const int ch = nt * 16 + lrow;
      float b2v = unc_b2[ch];
      for (int r = 0; r < 8; ++r) {
        float sig = sigmoidf(sg[nt][r] + b2v);
        int m = r + 8 * lhalf;
        out[(((size_t)(b * TT + t) * HWN) + row0 + m) * DD + ch] =
            y[nt][r] * sig;
      }
    }
    __syncthreads();
  }
}

// ======================================================================
extern "C" void kernel_launch(void* const* d_in, const int* in_sizes, int n_in,
                              void* d_out, int out_size, void* d_ws, size_t ws_size,
                              hipStream_t stream) {
  (void)in_sizes; (void)n_in; (void)out_size; (void)ws_size;
  const float* x           = (const float*)d_in[0];
  const float* dt_seq      = (const float*)d_in[1];
  const float* w_re        = (const float*)d_in[2];
  const float* w_im        = (const float*)d_in[3];
  const float* alpha_logit = (const float*)d_in[4];
  const float* lam_re      = (const float*)d_in[5];
  const float* lam_im      = (const float*)d_in[6];
  const float* h0_re       = (const float*)d_in[7];
  const float* h0_im       = (const float*)d_in[8];
  const float* ftd_re      = (const float*)d_in[9];
  const float* ftd_im      = (const float*)d_in[10];
  const float* ft_in_w     = (const float*)d_in[11];
  const float* ft_in_b     = (const float*)d_in[12];
  const float* ft_out_w    = (const float*)d_in[13];
  const float* ft_out_b    = (const float*)d_in[14];
  const float* ft_gc_w     = (const float*)d_in[15];
  const float* ft_gc_b     = (const float*)d_in[16];
  const float* ft_gate_w   = (const float*)d_in[17];
  const float* ft_gate_b   = (const float*)d_in[18];
  const float* ft_h0_re    = (const float*)d_in[19];
  const float* ft_h0_im    = (const float*)d_in[20];
  const float* unc_w1      = (const float*)d_in[21];
  const float* unc_b1      = (const float*)d_in[22];
  const float* unc_w2      = (const float*)d_in[23];
  const float* unc_b2      = (const float*)d_in[24];
  float* ws = (float*)d_ws;
  float* out = (float*)d_out;

  k1_xmean<<<BBN * TT, 256, 0, stream>>>(x, ws);
  k0_prepare<<<1, 256, 0, stream>>>(w_re, w_im, alpha_logit, lam_re, lam_im,
                                    ftd_re, ftd_im, dt_seq, unc_w1, unc_w2, ws);
  k2_flux<<<1, 256, 0, stream>>>(ws, ft_in_w, ft_in_b, ft_out_w, ft_out_b,
                                 ft_gc_w, ft_gc_b, ft_gate_w, ft_gate_b,
                                 ft_h0_re, ft_h0_im);
  k3_main<<<256, 64, 0, stream>>>(x, ws, h0_re, h0_im, unc_b1, unc_b2, out);
}